// clustering_dynamic_learning_45286135169473
// MI455X (gfx1250) — compile-verified
//
#include <hip/hip_runtime.h>
#include <hip/hip_bf16.h>

// ---------------- problem constants ----------------
#define B_ 8
#define N_ 325
#define K_ 16
#define C_ 8
#define D_ 64
#define EPS_ 1e-5f
#define MARGIN_ 0.5f
#define UPD_ 0.01f

// workspace layout (float offsets)
#define WS_SF 0                       // feat-BN scale per n      [325]
#define WS_TF 325                     // feat-BN shift per n      [325]
#define WS_SC 650                     // cent-BN scale per n      [325]
#define WS_TC 975                     // cent-BN shift per n      [325]
#define WS_CB 1300                    // CBb[n][c][j] = cent_bn@W1bot + b1 + tF*colsumTop   [325*8*64]
#define WS_LP (1300 + N_ * C_ * D_)   // per-n loss partial       [325]

#define OUT_UPDATED (B_ * N_ * C_ * D_)   // loss scalar lives at d_out[OUT_UPDATED]

typedef __attribute__((ext_vector_type(2))) float v2f;
typedef __attribute__((ext_vector_type(8))) float v8f;

// =====================================================================
// Kernel 1: per-n BN statistics -> per-n affine (scale, shift) for both
// the centroid BN (over C*D=512 values) and the feature BN (over the
// B*K*D=8192 gathered neighbor values).
// =====================================================================
__global__ void k_stats(const float* __restrict__ fushed,
                        const int* __restrict__ adj,
                        const float* __restrict__ cent,
                        const float* __restrict__ gC, const float* __restrict__ bC,
                        const float* __restrict__ gF, const float* __restrict__ bF,
                        float* __restrict__ ws) {
  const int n = blockIdx.x;
  const int tid = threadIdx.x; // 256 threads
  __shared__ float rs[256], rq[256];
  __shared__ int sIdx[B_ * K_]; // 128 gathered row indices for this n

  if (tid < B_ * K_) {
    int b = tid >> 4, k = tid & 15;
    sIdx[tid] = adj[(b * N_ + n) * K_ + k];
  }
  __syncthreads();

  // ---- feature stats: 8192 gathered values ----
  float s = 0.f, q = 0.f;
#pragma unroll
  for (int i = 0; i < 32; ++i) {
    int flat = i * 256 + tid;       // 0..8191
    int p = flat >> 6, d = flat & 63;
    int b = p >> 4;
    int row = sIdx[p];
    float v = fushed[(b * N_ + row) * D_ + d];
    s += v; q += v * v;
  }
  rs[tid] = s; rq[tid] = q;
  __syncthreads();
  for (int off = 128; off > 0; off >>= 1) {
    if (tid < off) { rs[tid] += rs[tid + off]; rq[tid] += rq[tid + off]; }
    __syncthreads();
  }
  if (tid == 0) {
    float m = rs[0] * (1.0f / 8192.0f);
    float v = rq[0] * (1.0f / 8192.0f) - m * m;
    float sc = gF[n] * rsqrtf(v + EPS_);
    ws[WS_SF + n] = sc;
    ws[WS_TF + n] = bF[n] - m * sc;
  }
  __syncthreads();

  // ---- centroid stats: 512 values ----
  s = 0.f; q = 0.f;
#pragma unroll
  for (int i = 0; i < 2; ++i) {
    float v = cent[n * (C_ * D_) + i * 256 + tid];
    s += v; q += v * v;
  }
  rs[tid] = s; rq[tid] = q;
  __syncthreads();
  for (int off = 128; off > 0; off >>= 1) {
    if (tid < off) { rs[tid] += rs[tid + off]; rq[tid] += rq[tid + off]; }
    __syncthreads();
  }
  if (tid == 0) {
    float m = rs[0] * (1.0f / 512.0f);
    float v = rq[0] * (1.0f / 512.0f) - m * m;
    float sc = gC[n] * rsqrtf(v + EPS_);
    ws[WS_SC + n] = sc;
    ws[WS_TC + n] = bC[n] - m * sc;
  }
}

// =====================================================================
// Kernel 2: CBb[n][c][j] = (cent_bn[n,c,:] @ W1_bot)[j] + b1[j]
//                          + tF[n] * colsumTop[j]
// (folds the feat-BN shift term t_F * sum_d W1_top[d][j] so the main
// kernel only needs a per-row scale on the WMMA result). 2600 rows; VALU.
// =====================================================================
__global__ void k_cb(const float* __restrict__ cent,
                     const float* __restrict__ W1,
                     const float* __restrict__ b1,
                     float* __restrict__ ws) {
  const int n = blockIdx.x;
  const int tid = threadIdx.x; // 256
  __shared__ float sc[C_ * D_];   // centroids[n] raw
  __shared__ float colT[D_], colB[D_];

  sc[tid] = cent[n * (C_ * D_) + tid];
  sc[tid + 256] = cent[n * (C_ * D_) + 256 + tid];
  if (tid < 64) {
    float st = 0.f, sb = 0.f;
#pragma unroll
    for (int d = 0; d < 64; ++d) {
      st += W1[d * 64 + tid];          // column sums of W1_top
      sb += W1[(64 + d) * 64 + tid];   // column sums of W1_bot
    }
    colT[tid] = st; colB[tid] = sb;
  }
  __syncthreads();

  const float sC = ws[WS_SC + n], tC = ws[WS_TC + n], tF = ws[WS_TF + n];
#pragma unroll
  for (int i = 0; i < 2; ++i) {
    int idx = i * 256 + tid;           // c*64 + j
    int c = idx >> 6, j = idx & 63;
    float acc = 0.f;
#pragma unroll
    for (int d = 0; d < 64; ++d) acc += sc[c * 64 + d] * W1[(64 + d) * 64 + j];
    ws[WS_CB + (n * C_ + c) * D_ + j] = sC * acc + tC * colB[j] + b1[j] + tF * colT[j];
  }
}

// =====================================================================
// Kernel 3 (fused main): one block per (b,n), 128 threads = 4 waves.
//   - gather 16 neighbor rows of feat + inp into LDS
//   - G(16x64) = rawFeat(16x64) @ W1_top(64x64) via V_WMMA_F32_16X16X4_F32
//     (wave w owns output cols [16w,16w+16); 16 chained k-steps)
//   - scores[k][c] = relu( relu(sF*G[k] + CBb[c]) . W2 + b2 )
//   - softmax over c, updated[b,n,c,t] = (1/K) sum_k simi * inp
// Fragment layouts per ISA 7.12.2:
//   A 16x4 f32 : lane L<16 -> row L, v0=K0,v1=K1 ; lane L+16 -> K2,K3
//   B  4x16 f32: assumed symmetric (lane&15 = col; half-wave picks K pair)
//   C/D 16x16  : vgpr v -> row v + 8*(lane>=16), col = lane&15
// =====================================================================
__global__ void __launch_bounds__(128)
k_main(const float* __restrict__ fushed,
       const float* __restrict__ inp,
       const int* __restrict__ adj,
       const float* __restrict__ W1,
       const float* __restrict__ W2,
       const float* __restrict__ b2p,
       const float* __restrict__ ws,
       float* __restrict__ out) {
  const int blk = blockIdx.x;
  const int b = blk / N_, n = blk % N_;
  const int tid = threadIdx.x; // 128

  __shared__ float sW1t[64 * 64];   // W1_top, row-major [d][j]
  __shared__ float sFeat[16 * 64];  // gathered raw feat rows
  __shared__ float sInp[16 * 64];   // gathered inp rows
  __shared__ float sG[16 * 64];     // WMMA result
  __shared__ float sCB[C_ * D_];    // CBb[n]
  __shared__ float sW2[64];
  __shared__ float sScore[16 * 8];
  __shared__ float sSimi[16 * 8];
  __shared__ int sIdx[16];

  if (tid < 16) sIdx[tid] = adj[(b * N_ + n) * K_ + tid];
  if (tid < 64) sW2[tid] = W2[tid];
#pragma unroll
  for (int i = 0; i < 32; ++i) { int idx = i * 128 + tid; sW1t[idx] = W1[idx]; }
#pragma unroll
  for (int i = 0; i < 4; ++i) { int idx = i * 128 + tid; sCB[idx] = ws[WS_CB + n * (C_ * D_) + idx]; }
  __syncthreads();

#pragma unroll
  for (int i = 0; i < 8; ++i) {
    int flat = i * 128 + tid;          // k*64 + d
    int k = flat >> 6, d = flat & 63;
    int row = sIdx[k];
    sFeat[flat] = fushed[(b * N_ + row) * D_ + d];
    sInp[flat]  = inp[(size_t)b * N_ * D_ + row * D_ + d];
  }
  __syncthreads();

  // ---- WMMA GEMM: G = rawFeat @ W1_top ----
  {
    const int w = tid >> 5;            // wave -> column tile
    const int lane = tid & 31;
    const int r = lane & 15;
    const int half = lane >> 4;        // 0: K+{0,1}, 1: K+{2,3}
    v8f acc = {};
#pragma unroll
    for (int kk = 0; kk < 16; ++kk) {
      const int kb = 4 * kk + 2 * half;
      v2f a; a.x = sFeat[r * 64 + kb];           a.y = sFeat[r * 64 + kb + 1];
      v2f bb; bb.x = sW1t[kb * 64 + 16 * w + r]; bb.y = sW1t[(kb + 1) * 64 + 16 * w + r];
      acc = __builtin_amdgcn_wmma_f32_16x16x4_f32(false, a, false, bb,
                                                  (short)0, acc, false, false);
    }
#pragma unroll
    for (int v = 0; v < 8; ++v) sG[(v + 8 * half) * 64 + 16 * w + r] = acc[v];
  }
  __syncthreads();

  // ---- per-(k,c) scores: 128 threads = 16x8 pairs ----
  const float sFn = ws[WS_SF + n];
  {
    const int k = tid >> 3, c = tid & 7;
    float acc = b2p[0];
#pragma unroll
    for (int j = 0; j < 64; ++j) {
      float h = fmaxf(sFn * sG[k * 64 + j] + sCB[c * 64 + j], 0.f);
      acc += h * sW2[j];
    }
    sScore[tid] = fmaxf(acc, 0.f);
  }
  __syncthreads();

  // ---- softmax over c (one thread per k) ----
  if (tid < 16) {
    float mx = -1e30f;
#pragma unroll
    for (int c = 0; c < 8; ++c) mx = fmaxf(mx, sScore[tid * 8 + c]);
    float e[8], sum = 0.f;
#pragma unroll
    for (int c = 0; c < 8; ++c) { e[c] = expf(sScore[tid * 8 + c] - mx); sum += e[c]; }
    float inv = 1.0f / sum;
#pragma unroll
    for (int c = 0; c < 8; ++c) sSimi[tid * 8 + c] = e[c] * inv;
  }
  __syncthreads();

  // ---- updated[b,n,c,t] = (1/K) sum_k simi[k,c] * inp[k,t] ----
#pragma unroll
  for (int i = 0; i < 4; ++i) {
    int idx = i * 128 + tid;           // c*64 + t
    int c = idx >> 6, t = idx & 63;
    float acc = 0.f;
#pragma unroll
    for (int k = 0; k < 16; ++k) acc += sSimi[k * 8 + c] * sInp[k * 64 + t];
    out[((size_t)(b * N_ + n) * C_ + c) * D_ + t] = acc * (1.0f / (float)K_);
  }
}

// =====================================================================
// Kernel 4: per-n margin loss on new_cent = 0.99*cent + 0.01*mean_b(updated)
// (_fast_cdist with x1==x2 reduces to ||xi||^2+||xj||^2-2<xi,xj> after
// centering by the per-n mean over C). Writes ws[WS_LP+n]; deterministic.
// =====================================================================
__global__ void k_loss(const float* __restrict__ cent,
                       const float* __restrict__ out,
                       float* __restrict__ ws) {
  const int n = blockIdx.x;
  const int tid = threadIdx.x; // 128
  __shared__ float snc[C_ * D_];
  __shared__ float sadj[D_];
  __shared__ float xn[C_];
  __shared__ float red[128];

#pragma unroll
  for (int i = 0; i < 4; ++i) {
    int idx = i * 128 + tid;           // c*64 + d
    float acc = 0.f;
#pragma unroll
    for (int b = 0; b < 8; ++b) acc += out[(size_t)(b * N_ + n) * (C_ * D_) + idx];
    snc[idx] = (1.0f - UPD_) * cent[n * (C_ * D_) + idx] + UPD_ * (acc * (1.0f / (float)B_));
  }
  __syncthreads();
  if (tid < 64) {
    float a = 0.f;
#pragma unroll
    for (int c = 0; c < 8; ++c) a += snc[c * 64 + tid];
    sadj[tid] = a * (1.0f / (float)C_);
  }
  __syncthreads();
#pragma unroll
  for (int i = 0; i < 4; ++i) { int idx = i * 128 + tid; snc[idx] -= sadj[idx & 63]; }
  __syncthreads();
  if (tid < 8) {
    float a = 0.f;
#pragma unroll
    for (int d = 0; d < 64; ++d) { float v = snc[tid * 64 + d]; a += v * v; }
    xn[tid] = a;
  }
  __syncthreads();

  float part = 0.f;
  if (tid < 64) {
    int c = tid >> 3, c2 = tid & 7;
    float dot = 0.f;
#pragma unroll
    for (int e = 0; e < 64; ++e) dot += snc[c * 64 + e] * snc[c2 * 64 + e];
    float res = fmaxf(xn[c] + xn[c2] - 2.0f * dot, 1e-30f);
    float dist = sqrtf(res);
    float target = (c == c2) ? 0.0f : MARGIN_;
    float r = fmaxf(target - dist, 0.0f);
    part = r * r;
  }
  red[tid] = part;
  __syncthreads();
  for (int off = 64; off > 0; off >>= 1) {
    if (tid < off) red[tid] += red[tid + off];
    __syncthreads();
  }
  if (tid == 0) ws[WS_LP + n] = red[0];
}

// =====================================================================
// Kernel 5: deterministic reduce of 325 per-n partials -> loss scalar.
// =====================================================================
__global__ void k_finalize(const float* __restrict__ ws, float* __restrict__ out) {
  __shared__ float red[512];
  const int tid = threadIdx.x; // 512
  red[tid] = (tid < N_) ? ws[WS_LP + tid] : 0.f;
  __syncthreads();
  for (int off = 256; off > 0; off >>= 1) {
    if (tid < off) red[tid] += red[tid + off];
    __syncthreads();
  }
  if (tid == 0) out[OUT_UPDATED] = red[0] * (1.0f / (float)N_);
}

extern "C" void kernel_launch(void* const* d_in, const int* in_sizes, int n_in,
                              void* d_out, int out_size, void* d_ws, size_t ws_size,
                              hipStream_t stream) {
  const float* fushed = (const float*)d_in[0];   // (B,N,D)
  const float* inp    = (const float*)d_in[1];   // (B,1,N,D)
  const int*   adj    = (const int*)d_in[2];     // (B,N,K)
  const float* cent   = (const float*)d_in[3];   // (N,C,D)
  const float* W1     = (const float*)d_in[4];   // (2D, D)
  const float* b1     = (const float*)d_in[5];   // (D,)
  const float* W2     = (const float*)d_in[6];   // (D,1)
  const float* b2     = (const float*)d_in[7];   // (1,)
  const float* gC     = (const float*)d_in[8];   // (N,)
  const float* bC     = (const float*)d_in[9];
  const float* gF     = (const float*)d_in[10];
  const float* bF     = (const float*)d_in[11];

  float* out = (float*)d_out;
  float* ws  = (float*)d_ws;

  k_stats<<<N_, 256, 0, stream>>>(fushed, adj, cent, gC, bC, gF, bF, ws);
  k_cb<<<N_, 256, 0, stream>>>(cent, W1, b1, ws);
  k_main<<<B_ * N_, 128, 0, stream>>>(fushed, inp, adj, W1, W2, b2, ws, out);
  k_loss<<<N_, 128, 0, stream>>>(cent, out, ws);
  k_finalize<<<1, 512, 0, stream>>>(ws, out);
}